// HGraph_8675833938445
// MI455X (gfx1250) — compile-verified
//
#include <hip/hip_runtime.h>
#include <cstdint>

typedef __attribute__((ext_vector_type(2))) float v2f;
typedef __attribute__((ext_vector_type(8))) float v8f;

#define N_NODES 90
#define M_PAD   96       // node rows padded to multiple of 16
#define F_IN    512
#define F_OUT   512
#define N_REL   4
#define BATCH   512

// Async global->LDS copy of 16B per lane (GVS mode: SGPR base + VGPR offset).
// ldsa = low 32 bits of generic LDS pointer (== byte offset in workgroup LDS).
#define ASYNC_LDS_B128(ldsa, goff, sbase)                                  \
    asm volatile("global_load_async_to_lds_b128 %0, %1, %2"                \
                 :: "v"(ldsa), "v"(goff), "s"(sbase) : "memory")
#define WAIT_ASYNC0() asm volatile("s_wait_asynccnt 0x0" ::: "memory")

__device__ __forceinline__ uint32_t lds_off(const void* p) {
    return (uint32_t)(uintptr_t)p;
}

// ---------------------------------------------------------------------------
// Kernel 0: Bsum[n,o] = sum_t B[t,n,o]
// ---------------------------------------------------------------------------
__global__ void bias_sum_kernel(const float* __restrict__ Bm,
                                float* __restrict__ Bsum) {
    int i = blockIdx.x * blockDim.x + threadIdx.x;
    const int sz = N_NODES * F_OUT;
    if (i < sz) {
        Bsum[i] = Bm[i] + Bm[sz + i] + Bm[2 * sz + i] + Bm[3 * sz + i];
    }
}

// ---------------------------------------------------------------------------
// Kernel 1: G[b] = (sum_r A[b,r] + I) @ F[b]   ([96pad x 512] per batch)
// One block per batch; A_sum resident in LDS; features streamed through LDS
// in 64-column slices via GLOBAL_LOAD_ASYNC_TO_LDS_B128 (ASYNCcnt-tracked).
// ---------------------------------------------------------------------------
#define AS_STRIDE 98     // 96x98 f32 = 37632 B
#define FS_COLS   64
#define FS_STRIDE 72     // 2 rows apart == 16 banks -> half-waves disjoint; 288B rows (16B mult)

__global__ __launch_bounds__(256) void
stage1_kernel(const float* __restrict__ feat, const float* __restrict__ A,
              float* __restrict__ G) {
    __shared__ __align__(16) float As[M_PAD * AS_STRIDE];   // 37632 B
    __shared__ __align__(16) float Fs[M_PAD * FS_STRIDE];   // 27648 B (65280 total)

    const int b    = blockIdx.x;
    const int tid  = threadIdx.x;
    const int lane = tid & 31;
    const int wave = tid >> 5;
    const int lm   = lane & 15;
    const int hi   = lane >> 4;       // 0: lanes 0-15, 1: lanes 16-31
    const int koff = hi * 2;          // upper half-wave holds K+2 in A/B frags

    // Zero-pad A_sum, then accumulate relations + identity.
    for (int i = tid; i < M_PAD * AS_STRIDE; i += 256) As[i] = 0.0f;
    // Zero the K-padding rows (90..95) of the feature slice once; the async
    // loads below only ever write rows 0..89.
    for (int i = tid; i < 6 * FS_COLS; i += 256) {
        int r = N_NODES + (i >> 6), c = i & 63;
        Fs[r * FS_STRIDE + c] = 0.0f;
    }
    __syncthreads();
    for (int i = tid; i < N_NODES * N_NODES; i += 256) {
        int n = i / N_NODES, m = i - n * N_NODES;
        const float* ap = A + (((size_t)b * N_REL) * N_NODES + n) * N_NODES + m;
        float s = ap[0] + ap[(size_t)N_NODES * N_NODES]
                + ap[(size_t)2 * N_NODES * N_NODES]
                + ap[(size_t)3 * N_NODES * N_NODES];
        if (n == m) s += 1.0f;
        As[n * AS_STRIDE + m] = s;
    }

    const char* fbase0 = (const char*)(feat + (size_t)b * N_NODES * F_IN);

    for (int ns = 0; ns < F_IN / FS_COLS; ++ns) {
        __syncthreads();   // previous slice fully consumed before overwrite
        // Async-stage feature slice rows 0..89 (90*16 = 1440 B128 lane-copies).
        const char* fbase = fbase0 + (size_t)ns * FS_COLS * 4;
        for (int i4 = tid; i4 < N_NODES * (FS_COLS / 4); i4 += 256) {
            int r = i4 >> 4, c4 = i4 & 15;
            uint32_t la = lds_off(&Fs[r * FS_STRIDE + c4 * 4]);
            uint32_t go = (uint32_t)((r * F_IN + c4 * 4) * 4);
            ASYNC_LDS_B128(la, go, fbase);
        }
        WAIT_ASYNC0();
        __syncthreads();

        // 24 output tiles, 3 per wave (same n0 -> shared B frags, 3 indep
        // accumulator chains). Uniform control flow around WMMA.
        for (int j = 0; j < 3; ++j) {
            int t  = wave + 8 * j;
            int m0 = (t >> 2) * 16;
            int n0 = (t & 3) * 16;
            v8f c = {};
            for (int k = 0; k < M_PAD; k += 4) {
                v2f a = *(const v2f*)&As[(m0 + lm) * AS_STRIDE + k + koff];
                v2f bb;
                bb.x = Fs[(k + koff) * FS_STRIDE + n0 + lm];
                bb.y = Fs[(k + koff + 1) * FS_STRIDE + n0 + lm];
                c = __builtin_amdgcn_wmma_f32_16x16x4_f32(
                        false, a, false, bb, (short)0, c, false, false);
            }
#pragma unroll
            for (int i = 0; i < 8; ++i) {
                int row = m0 + i + hi * 8;
                G[((size_t)b * M_PAD + row) * F_IN + ns * FS_COLS + n0 + lm] = c[i];
            }
        }
    }
}

// ---------------------------------------------------------------------------
// Kernel 2: out[b, rows_of_type_t, :] = G[b, rows] @ W[t] + Bsum[rows, :]
// Grid = (8 m-tiles, 512 batches); tiles never cross the 35/35/7/13 type
// partition. W and G k-slices async-staged through LDS; accumulators persist
// across the K loop; bias fused into the masked store.
// ---------------------------------------------------------------------------
#define WL_STRIDE 520    // 2 rows apart == 16 banks -> half-waves disjoint; 2080B rows
#define GS_STRIDE 20     // 80B rows (16B mult); 20*lm mod 64 distinct

__global__ __launch_bounds__(256) void
stage2_kernel(const float* __restrict__ G, const float* __restrict__ W,
              const float* __restrict__ Bsum, float* __restrict__ out) {
    __shared__ __align__(16) float Wl[16 * WL_STRIDE];   // 33280 B
    __shared__ __align__(16) float Gl[16 * GS_STRIDE];   // 1280 B

    const int mt   = blockIdx.x;   // 0..7
    const int b    = blockIdx.y;
    const int tid  = threadIdx.x;
    const int lane = tid & 31;
    const int wave = tid >> 5;
    const int lm   = lane & 15;
    const int hi   = lane >> 4;
    const int koff = hi * 2;

    static const int m0_tab[8]   = { 0, 16, 32, 35, 51, 67, 70, 77 };
    static const int rows_tab[8] = { 16, 16,  3, 16, 16,  3,  7, 13 };
    static const int type_tab[8] = { 0,  0,  0,  1,  1,  1,  2,  3 };
    const int m0    = m0_tab[mt];
    const int mrows = rows_tab[mt];
    const int ty    = type_tab[mt];

    const char* wbase = (const char*)(W + (size_t)ty * F_IN * F_OUT);
    const char* gbase = (const char*)(G + (size_t)b * M_PAD * F_IN);

    v8f acc[4] = {{}, {}, {}, {}};

    for (int kc = 0; kc < F_IN / 16; ++kc) {
        __syncthreads();   // previous slice fully consumed before overwrite
        // W k-slice: 16 x 512 f32 = 2048 B128 lane-copies (8 per thread).
        for (int i4 = tid; i4 < 16 * F_OUT / 4; i4 += 256) {
            int r = i4 >> 7, c4 = i4 & 127;
            uint32_t la = lds_off(&Wl[r * WL_STRIDE + c4 * 4]);
            uint32_t go = (uint32_t)((((kc * 16 + r) * F_OUT) + c4 * 4) * 4);
            ASYNC_LDS_B128(la, go, wbase);
        }
        // G tile slice: 16 x 16 f32 = 64 B128 lane-copies (threads 0..63).
        if (tid < 64) {
            int r = tid >> 2, c4 = tid & 3;
            uint32_t la = lds_off(&Gl[r * GS_STRIDE + c4 * 4]);
            uint32_t go = (uint32_t)(((m0 + r) * F_IN + kc * 16 + c4 * 4) * 4);
            ASYNC_LDS_B128(la, go, gbase);
        }
        WAIT_ASYNC0();
        __syncthreads();

#pragma unroll
        for (int nt = 0; nt < 4; ++nt) {
            const int n0 = wave * 64 + nt * 16;
#pragma unroll
            for (int k = 0; k < 16; k += 4) {
                v2f a = *(const v2f*)&Gl[lm * GS_STRIDE + k + koff];
                v2f bb;
                bb.x = Wl[(k + koff) * WL_STRIDE + n0 + lm];
                bb.y = Wl[(k + koff + 1) * WL_STRIDE + n0 + lm];
                acc[nt] = __builtin_amdgcn_wmma_f32_16x16x4_f32(
                              false, a, false, bb, (short)0, acc[nt], false, false);
            }
        }
    }

    // Fused bias + masked store (skip padding rows of partial tiles).
#pragma unroll
    for (int nt = 0; nt < 4; ++nt) {
        const int n0 = wave * 64 + nt * 16;
#pragma unroll
        for (int i = 0; i < 8; ++i) {
            int row = i + hi * 8;
            if (row < mrows) {
                int gm  = m0 + row;
                int col = n0 + lm;
                out[((size_t)b * N_NODES + gm) * F_OUT + col] =
                    acc[nt][i] + Bsum[gm * F_OUT + col];
            }
        }
    }
}

// ---------------------------------------------------------------------------
extern "C" void kernel_launch(void* const* d_in, const int* in_sizes, int n_in,
                              void* d_out, int out_size, void* d_ws, size_t ws_size,
                              hipStream_t stream) {
    (void)in_sizes; (void)n_in; (void)out_size; (void)ws_size;
    const float* feat = (const float*)d_in[0];   // [512, 90, 512]
    const float* A    = (const float*)d_in[1];   // [512, 4, 90, 90]
    const float* W    = (const float*)d_in[2];   // [4, 512, 512]
    const float* Bm   = (const float*)d_in[3];   // [4, 90, 512]
    float* out = (float*)d_out;                  // [512, 90, 512]

    // ws layout: G (512 x 96 x 512 f32 ~ 96 MB) | Bsum (90 x 512 f32)
    float* G    = (float*)d_ws;
    float* Bsum = G + (size_t)BATCH * M_PAD * F_IN;

    bias_sum_kernel<<<(N_NODES * F_OUT + 255) / 256, 256, 0, stream>>>(Bm, Bsum);
    stage1_kernel<<<BATCH, 256, 0, stream>>>(feat, A, G);
    stage2_kernel<<<dim3(8, BATCH), 256, 0, stream>>>(G, W, Bsum, out);
}